// MoE_OGB_83511344103772
// MI455X (gfx1250) — compile-verified
//
#include <hip/hip_runtime.h>
#include <math.h>

#define N_NODES   10000
#define NPAD      10016            // 313 * 32 : padded M for 32-row GEMM tiles
#define E_EDGES   320000
#define D_IN      128
#define D_HID     256
#define D_OUT     128
#define NUM_EXP   8
#define TOPK      4
#define COEF      0.001f

typedef __attribute__((ext_vector_type(2))) float v2f;
typedef __attribute__((ext_vector_type(8))) float v8f;

// ---------------------------------------------------------------------------
// Gating: logits = x @ w_gate ; top-4 softmax ; gates[N,8] ; importance/load
// ---------------------------------------------------------------------------
__global__ __launch_bounds__(256)
void gating_kernel(const float* __restrict__ x, const float* __restrict__ wg,
                   float* __restrict__ gates, float* __restrict__ importance,
                   float* __restrict__ loadv) {
    int n = blockIdx.x * blockDim.x + threadIdx.x;
    if (n >= N_NODES) return;
    float logit[NUM_EXP];
#pragma unroll
    for (int j = 0; j < NUM_EXP; j++) logit[j] = 0.f;
    const float* xr = x + (size_t)n * D_IN;
    for (int k = 0; k < D_IN; k++) {
        float xv = xr[k];
#pragma unroll
        for (int j = 0; j < NUM_EXP; j++)
            logit[j] = fmaf(xv, wg[k * NUM_EXP + j], logit[j]);
    }
    // top-4 (ties -> lower index, matching lax.top_k)
    float tmp[NUM_EXP];
#pragma unroll
    for (int j = 0; j < NUM_EXP; j++) tmp[j] = logit[j];
    int   idx[TOPK];
    float val[TOPK];
#pragma unroll
    for (int i = 0; i < TOPK; i++) {
        int bi = 0; float bv = tmp[0];
#pragma unroll
        for (int j = 1; j < NUM_EXP; j++)
            if (tmp[j] > bv) { bv = tmp[j]; bi = j; }
        idx[i] = bi; val[i] = bv; tmp[bi] = -INFINITY;
    }
    float m = val[0], s = 0.f, e[TOPK];
#pragma unroll
    for (int i = 0; i < TOPK; i++) { e[i] = expf(val[i] - m); s += e[i]; }
    float inv = 1.f / s;
    float g[NUM_EXP];
#pragma unroll
    for (int j = 0; j < NUM_EXP; j++) g[j] = 0.f;
#pragma unroll
    for (int i = 0; i < TOPK; i++) g[idx[i]] = e[i] * inv;
#pragma unroll
    for (int j = 0; j < NUM_EXP; j++) gates[(size_t)n * NUM_EXP + j] = g[j];
#pragma unroll
    for (int i = 0; i < TOPK; i++) {
        float gi = g[idx[i]];
        atomicAdd(&importance[idx[i]], gi);
        if (gi > 0.f) atomicAdd(&loadv[idx[i]], 1.f);
    }
}

// ---------------------------------------------------------------------------
// Degrees and symmetric edge normalization
// ---------------------------------------------------------------------------
__global__ __launch_bounds__(256)
void degree_kernel(const int* __restrict__ src, const int* __restrict__ dst,
                   float* __restrict__ deg_s, float* __restrict__ deg_d) {
    int e = blockIdx.x * blockDim.x + threadIdx.x;
    if (e >= E_EDGES) return;
    atomicAdd(&deg_s[src[e]], 1.f);
    atomicAdd(&deg_d[dst[e]], 1.f);
}

__global__ __launch_bounds__(256)
void edgew_kernel(const int* __restrict__ src, const int* __restrict__ dst,
                  const float* __restrict__ deg_s, const float* __restrict__ deg_d,
                  float* __restrict__ ew) {
    int e = blockIdx.x * blockDim.x + threadIdx.x;
    if (e >= E_EDGES) return;
    float a = rsqrtf(fmaxf(deg_s[src[e]], 1.f));
    float b = rsqrtf(fmaxf(deg_d[dst[e]], 1.f));
    ew[e] = a * b;
}

// ---------------------------------------------------------------------------
// float4 copy (x -> padded slab)
// ---------------------------------------------------------------------------
__global__ __launch_bounds__(256)
void copy4_kernel(const float4* __restrict__ src, float4* __restrict__ dst, long total4) {
    long gid = (long)blockIdx.x * blockDim.x + threadIdx.x;
    if (gid < total4) dst[gid] = src[gid];
}

// ---------------------------------------------------------------------------
// GEMM C[M,Ncols] = A[M,K] @ B[K,Ncols], f32, v_wmma_f32_16x16x4_f32.
// Each wave owns a 32x64 output tile (8 accumulators): per K-step it issues
// 2 A b64 loads + 8 B b32 loads and 8 WMMAs (1.25 vmem/wmma); A frags are
// reused 4x, B frags 2x.  M is padded to NPAD = 313*32.
// ISA layout (wave32):
//   A 16x4 : lanes 0-15 M=lane   {K+0,K+1}; lanes 16-31 M=lane-16 {K+2,K+3}
//   B 4x16 : lanes 0-15 N=lane   {K+0,K+1}; lanes 16-31 N=lane-16 {K+2,K+3}
//   C/D    : VGPR r -> row r (lanes<16) / row r+8 (lanes>=16), col = lane%16
// Launch: block = (32, Ncols/64), grid = (NPAD/32, 1).
// ---------------------------------------------------------------------------
__global__ __launch_bounds__(128)
void gemm_wmma_f32(const float* __restrict__ A, const float* __restrict__ B,
                   float* __restrict__ C, int K, int Ncols) {
    int lane = threadIdx.x;                       // 0..31
    int wave = threadIdx.y;                       // 0..Ncols/64-1
    int m0 = blockIdx.x * 32;
    int n0 = wave * 64;
    int half = lane >> 4;                         // 0 or 1
    int lid  = lane & 15;
    const float* arow0 = A + (size_t)(m0 + lid) * K + (half << 1);
    const float* arow1 = arow0 + (size_t)16 * K;
    const float* bbase = B + (size_t)(half << 1) * Ncols + (n0 + lid);
    v8f c00 = {}, c01 = {}, c02 = {}, c03 = {};
    v8f c10 = {}, c11 = {}, c12 = {}, c13 = {};
    for (int k = 0; k < K; k += 4) {
        v2f a0 = *(const v2f*)(arow0 + k);        // rows m0+lid    {K+2h,K+2h+1}
        v2f a1 = *(const v2f*)(arow1 + k);        // rows m0+16+lid
        const float* bk = bbase + (size_t)k * Ncols;
        v2f b0, b1, b2, b3;
        b0.x = bk[0];       b0.y = bk[Ncols];
        b1.x = bk[16];      b1.y = bk[16 + Ncols];
        b2.x = bk[32];      b2.y = bk[32 + Ncols];
        b3.x = bk[48];      b3.y = bk[48 + Ncols];
        c00 = __builtin_amdgcn_wmma_f32_16x16x4_f32(false, a0, false, b0, (short)0, c00, false, false);
        c01 = __builtin_amdgcn_wmma_f32_16x16x4_f32(false, a0, false, b1, (short)0, c01, false, false);
        c02 = __builtin_amdgcn_wmma_f32_16x16x4_f32(false, a0, false, b2, (short)0, c02, false, false);
        c03 = __builtin_amdgcn_wmma_f32_16x16x4_f32(false, a0, false, b3, (short)0, c03, false, false);
        c10 = __builtin_amdgcn_wmma_f32_16x16x4_f32(false, a1, false, b0, (short)0, c10, false, false);
        c11 = __builtin_amdgcn_wmma_f32_16x16x4_f32(false, a1, false, b1, (short)0, c11, false, false);
        c12 = __builtin_amdgcn_wmma_f32_16x16x4_f32(false, a1, false, b2, (short)0, c12, false, false);
        c13 = __builtin_amdgcn_wmma_f32_16x16x4_f32(false, a1, false, b3, (short)0, c13, false, false);
    }
    float* crow0 = C + (size_t)(m0 + (half << 3)) * Ncols + (n0 + lid);
    float* crow1 = crow0 + (size_t)16 * Ncols;
#pragma unroll
    for (int r = 0; r < 8; r++) {
        size_t o = (size_t)r * Ncols;
        crow0[o]      = c00[r];
        crow0[o + 16] = c01[r];
        crow0[o + 32] = c02[r];
        crow0[o + 48] = c03[r];
        crow1[o]      = c10[r];
        crow1[o + 16] = c11[r];
        crow1[o + 32] = c12[r];
        crow1[o + 48] = c13[r];
    }
}

// ---------------------------------------------------------------------------
// Edge aggregation: agg[dst] += h[src] * ew[e]   (float4 gather, f32 atomics)
// 1<<shift float4 lanes per edge (shift=6 for d=256, 5 for d=128)
// ---------------------------------------------------------------------------
__global__ __launch_bounds__(256)
void aggregate_kernel(const float* __restrict__ h, const int* __restrict__ src,
                      const int* __restrict__ dst, const float* __restrict__ ew,
                      float* __restrict__ agg, int d, int shift) {
    long gid = (long)blockIdx.x * blockDim.x + threadIdx.x;
    int  e   = (int)(gid >> shift);
    int  f4  = (int)(gid & ((1 << shift) - 1));
    if (e >= E_EDGES) return;
    int   s = src[e], t = dst[e];
    float w = ew[e];
    const float4* hp = (const float4*)(h + (size_t)s * d) + f4;
    float4 v = *hp;
    float* ap = agg + (size_t)t * d + ((size_t)f4 << 2);
    atomicAdd(ap + 0, v.x * w);
    atomicAdd(ap + 1, v.y * w);
    atomicAdd(ap + 2, v.z * w);
    atomicAdd(ap + 3, v.w * w);
}

// ---------------------------------------------------------------------------
// out = relu(agg + b)   (hidden layers)
// ---------------------------------------------------------------------------
__global__ __launch_bounds__(256)
void bias_relu_kernel(const float* __restrict__ agg, const float* __restrict__ b,
                      float* __restrict__ out, int mask, long total) {
    long gid = (long)blockIdx.x * blockDim.x + threadIdx.x;
    if (gid >= total) return;
    int f = (int)(gid & mask);
    out[gid] = fmaxf(agg[gid] + b[f], 0.f);
}

// ---------------------------------------------------------------------------
// y += gate[n,expert] * (agg + b)   (last layer of each expert)
// ---------------------------------------------------------------------------
__global__ __launch_bounds__(256)
void out_accum_kernel(const float* __restrict__ agg, const float* __restrict__ b,
                      const float* __restrict__ gates, int expert,
                      float* __restrict__ y) {
    long gid = (long)blockIdx.x * blockDim.x + threadIdx.x;
    if (gid >= (long)N_NODES * D_OUT) return;
    int n = (int)(gid >> 7);          // D_OUT == 128
    int f = (int)(gid & 127);
    float g = gates[(size_t)n * NUM_EXP + expert];
    y[gid] = fmaf(g, agg[gid] + b[f], y[gid]);
}

// ---------------------------------------------------------------------------
// loss = (cv2(importance) + cv2(load)) * COEF  (ddof=1 variance over 8 vals)
// ---------------------------------------------------------------------------
__global__ void loss_kernel(const float* __restrict__ imp,
                            const float* __restrict__ loadv,
                            float* __restrict__ out_loss) {
    if (threadIdx.x != 0 || blockIdx.x != 0) return;
    float loss = 0.f;
    for (int which = 0; which < 2; which++) {
        const float* v = which ? loadv : imp;
        float mean = 0.f;
        for (int j = 0; j < NUM_EXP; j++) mean += v[j];
        mean *= (1.f / NUM_EXP);
        float var = 0.f;
        for (int j = 0; j < NUM_EXP; j++) {
            float d = v[j] - mean;
            var += d * d;
        }
        var *= (1.f / (NUM_EXP - 1));
        loss += var / (mean * mean + 1e-10f);
    }
    out_loss[0] = loss * COEF;
}

// ---------------------------------------------------------------------------
extern "C" void kernel_launch(void* const* d_in, const int* in_sizes, int n_in,
                              void* d_out, int out_size, void* d_ws, size_t ws_size,
                              hipStream_t stream) {
    (void)in_sizes; (void)n_in; (void)out_size; (void)ws_size;
    const float* x    = (const float*)d_in[0];
    const int*   esrc = (const int*)  d_in[1];
    const int*   edst = (const int*)  d_in[2];
    const float* wg   = (const float*)d_in[3];

    // ---- workspace partition (1KB-aligned slabs) ----
    float* ws = (float*)d_ws;
    size_t off = 0;
    auto alloc = [&](size_t n) { float* p = ws + off; off += (n + 255) & ~(size_t)255; return p; };
    float* gates      = alloc((size_t)N_NODES * NUM_EXP);
    float* importance = alloc(NUM_EXP);
    float* loadv      = alloc(NUM_EXP);
    float* deg_s      = alloc(N_NODES);
    float* deg_d      = alloc(N_NODES);
    float* ew         = alloc(E_EDGES);
    float* xpad       = alloc((size_t)NPAD * D_IN);       // padded copy of x
    float* hGemm      = alloc((size_t)NPAD * D_HID);      // GEMM output (padded)
    float* hAgg       = alloc((size_t)NPAD * D_HID);      // scatter accumulator
    float* hAct       = alloc((size_t)NPAD * D_HID);      // post-activation (padded)

    float* y = (float*)d_out;                              // [N*D_OUT] then loss

    hipMemsetAsync(y,          0, sizeof(float) * (size_t)N_NODES * D_OUT, stream);
    hipMemsetAsync(importance, 0, sizeof(float) * NUM_EXP, stream);
    hipMemsetAsync(loadv,      0, sizeof(float) * NUM_EXP, stream);
    hipMemsetAsync(deg_s,      0, sizeof(float) * N_NODES, stream);
    hipMemsetAsync(deg_d,      0, sizeof(float) * N_NODES, stream);
    // zero the pad rows of the GEMM A-side buffers (done once; pad rows of
    // hAct are never written by bias_relu, pad rows of xpad never by copy4)
    hipMemsetAsync(xpad + (size_t)N_NODES * D_IN,  0,
                   sizeof(float) * (size_t)(NPAD - N_NODES) * D_IN,  stream);
    hipMemsetAsync(hAct + (size_t)N_NODES * D_HID, 0,
                   sizeof(float) * (size_t)(NPAD - N_NODES) * D_HID, stream);

    long x4 = (long)N_NODES * D_IN / 4;
    copy4_kernel<<<(unsigned)((x4 + 255) / 256), 256, 0, stream>>>(
        (const float4*)x, (float4*)xpad, x4);

    gating_kernel<<<(N_NODES + 255) / 256, 256, 0, stream>>>(x, wg, gates, importance, loadv);
    degree_kernel<<<(E_EDGES + 255) / 256, 256, 0, stream>>>(esrc, edst, deg_s, deg_d);
    edgew_kernel <<<(E_EDGES + 255) / 256, 256, 0, stream>>>(esrc, edst, deg_s, deg_d, ew);
    loss_kernel  <<<1, 32, 0, stream>>>(importance, loadv, y + (size_t)N_NODES * D_OUT);

    // ---- experts: expert e has 2+e layers; params flattened W,b per layer ----
    int pi = 4;
    for (int e = 0; e < NUM_EXP; e++) {
        int L = 2 + e;
        const float* cur = xpad;
        int din = D_IN;
        for (int l = 0; l < L; l++) {
            int dout = (l == L - 1) ? D_OUT : D_HID;
            const float* W = (const float*)d_in[pi++];
            const float* b = (const float*)d_in[pi++];

            // h = cur @ W   (NPAD x din @ din x dout); 32-row stripes
            dim3 gg(NPAD / 32, 1);
            dim3 gb(32, dout / 64);
            gemm_wmma_f32<<<gg, gb, 0, stream>>>(cur, W, hGemm, din, dout);

            // agg = segment_sum(h[src] * ew, dst)   (only real nodes touched)
            hipMemsetAsync(hAgg, 0, sizeof(float) * (size_t)N_NODES * dout, stream);
            int shift = (dout == 256) ? 6 : 5;
            long tot = (long)E_EDGES << shift;
            aggregate_kernel<<<(unsigned)((tot + 255) / 256), 256, 0, stream>>>(
                hGemm, esrc, edst, ew, hAgg, dout, shift);

            if (l < L - 1) {
                long te = (long)N_NODES * dout;
                bias_relu_kernel<<<(unsigned)((te + 255) / 256), 256, 0, stream>>>(
                    hAgg, b, hAct, dout - 1, te);
                cur = hAct;
                din = dout;
            } else {
                long te = (long)N_NODES * D_OUT;
                out_accum_kernel<<<(unsigned)((te + 255) / 256), 256, 0, stream>>>(
                    hAgg, b, gates, e, y);
            }
        }
    }
}